// DropNorm_17008070492698
// MI455X (gfx1250) — compile-verified
//
#include <hip/hip_runtime.h>
#include <hip/hip_bf16.h>
#include <math.h>

typedef float v4f __attribute__((ext_vector_type(4)));
typedef float v2f __attribute__((ext_vector_type(2)));
typedef float v8f __attribute__((ext_vector_type(8)));

#define NUMEL      (16LL * 2048LL * 1024LL)     // 33,554,432
#define N4         (NUMEL / 4)                  // 8,388,608 float4s
#define NNZ_INV    (1.0f / 16777216.0f)         // exact 2^-24
#define EPSV       1e-5f

#define RED_BLOCKS  2048
#define RED_THREADS 256
#define EW_BLOCKS   8192
#define EW_THREADS  256

// ---------------------------------------------------------------------------
// Wave32 sum via V_WMMA_F32_16X16X4_F32.
// A (16x4 f32, 2 VGPRs): VGPR0 = per-lane partial acc  (A[m][0]=acc[m],
// A[m][2]=acc[m+16]), VGPR1 = 0. B (4x16) = all ones.
// D[m][n] = acc[m] + acc[m+16]  (replicated over n).
// Sum of the 8 D VGPRs gives rows 0..7 in lanes 0-15 and rows 8..15 in
// lanes 16-31; one shfl_xor(16) add produces the full 32-lane total.
// EXEC is all-1s at every call site (uniform trip counts, no divergence).
// ---------------------------------------------------------------------------
__device__ __forceinline__ float wave_sum_wmma(float acc) {
    v2f a; a.x = acc;  a.y = 0.0f;
    v2f b; b.x = 1.0f; b.y = 1.0f;
    v8f c = {};
    v8f d = __builtin_amdgcn_wmma_f32_16x16x4_f32(
        /*neg_a=*/false, a, /*neg_b=*/false, b,
        /*c_mod=*/(short)0, c, /*reuse_a=*/false, /*reuse_b=*/false);
    float s = d[0] + d[1] + d[2] + d[3] + d[4] + d[5] + d[6] + d[7];
    s += __shfl_xor(s, 16, 32);   // combine the two half-wave row groups
    return s;                     // every lane now holds the wave total
}

__device__ __forceinline__ float block_sum(float v, float* lds) {
    float w = wave_sum_wmma(v);
    const int lane = threadIdx.x & 31;
    const int wid  = threadIdx.x >> 5;
    if (lane == 0) lds[wid] = w;
    __syncthreads();
    float total = 0.0f;
    if (threadIdx.x == 0) {
        const int nw = blockDim.x >> 5;
        for (int i = 0; i < nw; ++i) total += lds[i];   // fixed order
    }
    return total;   // valid in thread 0 only
}

// ---------------------------------------------------------------------------
// Pass 1 / Pass 2 reduction. phase 0: partial += x*mask
//                            phase 1: partial += |x*mask - mu| * mask
// x loaded with regular temporal hint (128MB fits the 192MB L2 for re-reads);
// mask streamed non-temporally.
// ---------------------------------------------------------------------------
__global__ __launch_bounds__(RED_THREADS)
void reduce_kernel(const float* __restrict__ x, const float* __restrict__ mask,
                   float* __restrict__ partials, const float* __restrict__ muPtr,
                   int phase) {
    __shared__ float lds[RED_THREADS / 32];

    const long long tid    = (long long)blockIdx.x * blockDim.x + threadIdx.x;
    const long long stride = (long long)gridDim.x * blockDim.x;
    const float mu = (phase == 1) ? muPtr[0] : 0.0f;

    const v4f* x4 = (const v4f*)x;
    const v4f* m4 = (const v4f*)mask;

    v4f acc = {0.0f, 0.0f, 0.0f, 0.0f};
    for (long long i = tid; i < N4; i += stride) {
        v4f xv = x4[i];                                // temporal (L2-resident)
        v4f mv = __builtin_nontemporal_load(&m4[i]);   // streamed
        v4f xm = xv * mv;
        if (phase == 0) {
            acc += xm;
        } else {
            acc.x += fabsf(xm.x - mu) * mv.x;
            acc.y += fabsf(xm.y - mu) * mv.y;
            acc.z += fabsf(xm.z - mu) * mv.z;
            acc.w += fabsf(xm.w - mu) * mv.w;
        }
    }

    float s = (acc.x + acc.y) + (acc.z + acc.w);
    float total = block_sum(s, lds);
    if (threadIdx.x == 0) partials[blockIdx.x] = total;
}

// Second-level reduction over RED_BLOCKS partials (single block, fixed order).
__global__ __launch_bounds__(RED_THREADS)
void finalize_reduce(const float* __restrict__ partials, float* __restrict__ dst) {
    __shared__ float lds[RED_THREADS];
    float s = 0.0f;
    for (int i = threadIdx.x; i < RED_BLOCKS; i += RED_THREADS) s += partials[i];
    lds[threadIdx.x] = s;
    __syncthreads();
    for (int off = RED_THREADS / 2; off > 0; off >>= 1) {
        if (threadIdx.x < off) lds[threadIdx.x] += lds[threadIdx.x + off];
        __syncthreads();
    }
    if (threadIdx.x == 0) dst[0] = lds[0] * NNZ_INV;
}

// ---------------------------------------------------------------------------
// Elementwise: out = gamma * (x*mask - mu) * rstd + beta
// ---------------------------------------------------------------------------
__global__ __launch_bounds__(EW_THREADS)
void normalize_kernel(const float* __restrict__ x, const float* __restrict__ mask,
                      const float* __restrict__ gamma, const float* __restrict__ beta,
                      float* __restrict__ out, const float* __restrict__ stats) {
    const float mu   = stats[0];
    const float rstd = 1.0f / sqrtf(stats[1] + EPSV);

    const long long tid    = (long long)blockIdx.x * blockDim.x + threadIdx.x;
    const long long stride = (long long)gridDim.x * blockDim.x;

    const v4f* x4 = (const v4f*)x;
    const v4f* m4 = (const v4f*)mask;
    const v4f* g4 = (const v4f*)gamma;
    const v4f* b4 = (const v4f*)beta;
    v4f*       o4 = (v4f*)out;

    for (long long i = tid; i < N4; i += stride) {
        v4f xv = x4[i];                                // may hit L2 from pass 2
        v4f mv = __builtin_nontemporal_load(&m4[i]);
        v4f gv = __builtin_nontemporal_load(&g4[i]);
        v4f bv = __builtin_nontemporal_load(&b4[i]);
        v4f xh = (xv * mv - mu) * rstd;
        v4f ov = gv * xh + bv;
        __builtin_nontemporal_store(ov, &o4[i]);
    }
}

// ---------------------------------------------------------------------------
extern "C" void kernel_launch(void* const* d_in, const int* in_sizes, int n_in,
                              void* d_out, int out_size, void* d_ws, size_t ws_size,
                              hipStream_t stream) {
    const float* x     = (const float*)d_in[0];
    const float* mask  = (const float*)d_in[1];
    const float* gamma = (const float*)d_in[2];
    const float* beta  = (const float*)d_in[3];
    float*       out   = (float*)d_out;

    float* stats    = (float*)d_ws;   // [0] = mu, [1] = sigma2 (mean abs dev)
    float* partials = stats + 2;      // RED_BLOCKS floats

    // Pass 1: mu
    reduce_kernel<<<RED_BLOCKS, RED_THREADS, 0, stream>>>(x, mask, partials, stats, 0);
    finalize_reduce<<<1, RED_THREADS, 0, stream>>>(partials, &stats[0]);
    // Pass 2: mean absolute deviation
    reduce_kernel<<<RED_BLOCKS, RED_THREADS, 0, stream>>>(x, mask, partials, stats, 1);
    finalize_reduce<<<1, RED_THREADS, 0, stream>>>(partials, &stats[1]);
    // Pass 3: normalize
    normalize_kernel<<<EW_BLOCKS, EW_THREADS, 0, stream>>>(x, mask, gamma, beta, out, stats);
}